// RNNClassifier_7945689498052
// MI455X (gfx1250) — compile-verified
//
#include <hip/hip_runtime.h>
#include <math.h>

typedef __attribute__((ext_vector_type(2))) float v2f;
typedef __attribute__((ext_vector_type(8))) float v8f;

#define B_ 256
#define T_ 1024
#define D_ 128
#define H_ 128

// ---------------------------------------------------------------------------
// Phase 1: xp[t][b][h] = x[b][t][:] . W_ih[h][:] + b_ih[h] + b_hh[h]
// One block = one (b, t0..t0+15) M-tile; 8 waves cover the 8 N-tiles of H=128.
// fp32 WMMA 16x16x4, two independent accumulator chains over K=128.
// ---------------------------------------------------------------------------
__global__ __launch_bounds__(256) void rnn_inproj(
    const float* __restrict__ x, const float* __restrict__ W_ih,
    const float* __restrict__ b_ih, const float* __restrict__ b_hh,
    float* __restrict__ xp)
{
  __shared__ float xs[16][132];            // padded: stride 132 -> bank-clean
  const int tid  = threadIdx.x;
  const int lane = tid & 31;
  const int wave = tid >> 5;               // N tile 0..7
  const int blk  = blockIdx.x;             // 0..16383
  const int b    = blk >> 6;               // 64 t-tiles per batch row
  const int t0   = (blk & 63) << 4;

  // Stage x[b][t0..t0+15][0..127] into LDS (coalesced, read-once -> NT).
  {
    const float* src = x + ((size_t)b * T_ + t0) * D_;
    for (int i = tid; i < 16 * D_; i += 256) {
      int r = i >> 7, c2 = i & 127;
      xs[r][c2] = __builtin_nontemporal_load(&src[r * D_ + c2]);
    }
  }

  const int n0   = wave << 4;
  const int lrow = lane & 15;
  const int khi  = (lane >> 4) << 1;       // K sub-offset: 0 (lo lanes) / 2 (hi)
  const int mhi  = (lane >> 4) << 3;       // C-matrix M offset: 0 / 8

  // Preload B fragments in registers: B[k][n] = W_ih[n0+n][k]
  v2f bfrag[32];
  {
    const float* wr = W_ih + (size_t)(n0 + lrow) * D_;
#pragma unroll
    for (int kk = 0; kk < 32; ++kk) {
      int kb = (kk << 2) + khi;
      bfrag[kk].x = wr[kb];
      bfrag[kk].y = wr[kb + 1];
    }
  }
  __syncthreads();

  v8f c0 = {};
  v8f c1 = {};
#pragma unroll
  for (int kk = 0; kk < 32; kk += 2) {
    int kb0 = (kk << 2) + khi;
    int kb1 = kb0 + 4;
    v2f a0, a1;
    a0.x = xs[lrow][kb0];
    a0.y = xs[lrow][kb0 + 1];
    a1.x = xs[lrow][kb1];
    a1.y = xs[lrow][kb1 + 1];
    c0 = __builtin_amdgcn_wmma_f32_16x16x4_f32(false, a0, false, bfrag[kk],
                                               (short)0, c0, false, false);
    c1 = __builtin_amdgcn_wmma_f32_16x16x4_f32(false, a1, false, bfrag[kk + 1],
                                               (short)0, c1, false, false);
  }

  const float bias = b_ih[n0 + lrow] + b_hh[n0 + lrow];
#pragma unroll
  for (int r = 0; r < 8; ++r) {
    int t = t0 + r + mhi;                  // C row M = r + 8*(lane>=16)
    xp[((size_t)t * B_ + b) * H_ + n0 + lrow] = c0[r] + c1[r] + bias;
  }
}

// ---------------------------------------------------------------------------
// Phase 2: sequential scan. One block per 16-row batch strip (16 blocks).
// Double-buffered h strip in LDS (1 barrier/step); each wave holds its
// 128x16 W_hh^T slice in 64 VGPRs for all 1024 steps; xp loads for t+1
// pipelined behind the WMMAs of t. Tail: fused FC(128->2) + sigmoid.
// ---------------------------------------------------------------------------
__global__ __launch_bounds__(256) void rnn_scan(
    const float* __restrict__ xp, const float* __restrict__ W_hh,
    const float* __restrict__ W_fc, const float* __restrict__ b_fc,
    float* __restrict__ out)
{
  __shared__ float hs[2][16][132];
  const int tid  = threadIdx.x;
  const int lane = tid & 31;
  const int wave = tid >> 5;
  const int b0   = blockIdx.x << 4;

  const int n0   = wave << 4;
  const int lrow = lane & 15;
  const int khi  = (lane >> 4) << 1;
  const int mhi  = (lane >> 4) << 3;

  // Persistent B fragments: B[k][n] = W_hh[n0+n][k]
  v2f bfrag[32];
  {
    const float* wr = W_hh + (size_t)(n0 + lrow) * H_;
#pragma unroll
    for (int kk = 0; kk < 32; ++kk) {
      int kb = (kk << 2) + khi;
      bfrag[kk].x = wr[kb];
      bfrag[kk].y = wr[kb + 1];
    }
  }

  for (int i = tid; i < 16 * 132; i += 256) (&hs[0][0][0])[i] = 0.0f;

  // Preload xp tile for t=0 (8 lanes-worth of C-layout gathers, NT: read-once)
  const float* pbase = xp + (size_t)(b0 + mhi) * H_ + n0 + lrow;
  v8f cn;
#pragma unroll
  for (int r = 0; r < 8; ++r)
    cn[r] = __builtin_nontemporal_load(&pbase[(size_t)r * H_]);

  __syncthreads();

  for (int t = 0; t < T_; ++t) {
    v8f a0 = cn;                           // chain 0 starts from xp tile
    v8f a1 = {};                           // chain 1 starts from zero

    // Software pipeline: issue next step's xp loads now (independent of LDS).
    {
      size_t tn = (size_t)((t + 1 < T_) ? t + 1 : t) * B_ * H_;
#pragma unroll
      for (int r = 0; r < 8; ++r)
        cn[r] = __builtin_nontemporal_load(&pbase[tn + (size_t)r * H_]);
    }

    const float (*hc)[132] = hs[t & 1];
#pragma unroll
    for (int kk = 0; kk < 32; kk += 2) {
      int kb0 = (kk << 2) + khi;
      int kb1 = kb0 + 4;
      v2f f0, f1;
      f0.x = hc[lrow][kb0];
      f0.y = hc[lrow][kb0 + 1];
      f1.x = hc[lrow][kb1];
      f1.y = hc[lrow][kb1 + 1];
      a0 = __builtin_amdgcn_wmma_f32_16x16x4_f32(false, f0, false, bfrag[kk],
                                                 (short)0, a0, false, false);
      a1 = __builtin_amdgcn_wmma_f32_16x16x4_f32(false, f1, false, bfrag[kk + 1],
                                                 (short)0, a1, false, false);
    }

    float (*hn)[132] = hs[(t & 1) ^ 1];
#pragma unroll
    for (int r = 0; r < 8; ++r)
      hn[r + mhi][n0 + lrow] = tanhf(a0[r] + a1[r]);

    __syncthreads();                       // publish new h / retire old reads
  }

  // Final h is in hs[T_ & 1] == hs[0].
  // Fused classifier head: out[b][o] = sigmoid(h[b] . W_fc[o] + b_fc[o])
  if (tid < 32) {
    int o = tid & 1, m = tid >> 1;
    float acc = b_fc[o];
    const float* wf = W_fc + o * H_;
#pragma unroll 8
    for (int d = 0; d < H_; ++d) acc += hs[0][m][d] * wf[d];
    out[(b0 + m) * 2 + o] = 1.0f / (1.0f + expf(-acc));
  }
}

// ---------------------------------------------------------------------------
extern "C" void kernel_launch(void* const* d_in, const int* in_sizes, int n_in,
                              void* d_out, int out_size, void* d_ws, size_t ws_size,
                              hipStream_t stream) {
  const float* x    = (const float*)d_in[0];
  const float* W_ih = (const float*)d_in[1];
  const float* W_hh = (const float*)d_in[2];
  const float* b_ih = (const float*)d_in[3];
  const float* b_hh = (const float*)d_in[4];
  const float* W_fc = (const float*)d_in[5];
  const float* b_fc = (const float*)d_in[6];

  float* xp  = (float*)d_ws;               // T*B*H fp32 = 128 MiB scratch
  float* out = (float*)d_out;

  rnn_inproj<<<dim3((B_ * T_) / 16), dim3(256), 0, stream>>>(x, W_ih, b_ih, b_hh, xp);
  rnn_scan<<<dim3(B_ / 16), dim3(256), 0, stream>>>(xp, W_hh, W_fc, b_fc, out);
}